// MatchLoss_13657996001888
// MI455X (gfx1250) — compile-verified
//
#include <hip/hip_runtime.h>
#include <hip/hip_bf16.h>

#define MATCHLOSS_EPS 1e-8f

typedef __attribute__((ext_vector_type(2))) float v2f;
typedef __attribute__((ext_vector_type(8))) float v8f;

// Single-workgroup fused reduction:
//   u  = log(d+eps) - log(t+eps)
//   S1 = sum(u), S2 = sum(u*u), S3 = sum((d-t)^2)
//   out = 2*(S2/N - (S1/N)^2) + S3/N
// 128 KB of input => DRAM time ~6 ns at 23.3 TB/s: latency-bound, so one
// block / one pass / zero atomics. Cross-wave combine uses exact-f32
// V_WMMA_F32_16X16X4_F32 with an all-ones B matrix as a matrix-pipe reducer.
__global__ __launch_bounds__(1024) void matchloss_kernel(
    const float* __restrict__ dptr, const float* __restrict__ tptr,
    float* __restrict__ out, int N) {
  const int tid  = threadIdx.x;       // 0..1023
  const int lane = tid & 31;          // wave32
  const int wave = tid >> 5;          // 0..31

  float s1 = 0.0f, s2 = 0.0f, s3 = 0.0f;

  // Coalesced float4 loads: thread tid handles elements [4*tid .. ) stride 4096.
  const float4* d4 = reinterpret_cast<const float4*>(dptr);
  const float4* t4 = reinterpret_cast<const float4*>(tptr);
  const int n4 = N >> 2;
  for (int i = tid; i < n4; i += 1024) {
    float4 dv = d4[i];
    float4 tv = t4[i];
    float de[4] = {dv.x, dv.y, dv.z, dv.w};
    float te[4] = {tv.x, tv.y, tv.z, tv.w};
#pragma unroll
    for (int j = 0; j < 4; ++j) {
      float u  = __builtin_logf(de[j] + MATCHLOSS_EPS) -
                 __builtin_logf(te[j] + MATCHLOSS_EPS);
      float dd = de[j] - te[j];
      s1 += u;
      s2 += u * u;
      s3 += dd * dd;
    }
  }

  // Intra-wave (32-lane) butterfly reduction.
#pragma unroll
  for (int off = 16; off > 0; off >>= 1) {
    s1 += __shfl_xor(s1, off, 32);
    s2 += __shfl_xor(s2, off, 32);
    s3 += __shfl_xor(s3, off, 32);
  }

  __shared__ float p1[32];
  __shared__ float p2[32];
  __shared__ float p3[32];
  if (lane == 0) {
    p1[wave] = s1;
    p2[wave] = s2;
    p3[wave] = s3;
  }
  __syncthreads();

  // Cross-wave combine via exact-f32 WMMA, executed uniformly by every wave
  // (EXEC must be all-ones for WMMA; branch-free construction below).
  //
  // A-matrix 16x4 f32 layout (ISA 7.12.2): lane L holds row m = L%16;
  //   VGPR0 = A[m, 2*(L/16)], VGPR1 = A[m, 2*(L/16)+1].
  // Pack: rows 0..7  <- p1[ m*4 + k ]   (32 values of S1-partials)
  //       rows 8..15 <- p2[(m-8)*4 + k] (32 values of S2-partials)
  // B = all ones (layout irrelevant: constant matrix).
  // D[m,n] = sum_k A[m,k]; summing the 8 C/D VGPRs yields
  //   lane 0  -> sum rows 0..7  = S1
  //   lane 16 -> sum rows 8..15 = S2
  const int m  = lane & 15;
  const int kb = (lane >> 4) << 1;       // 0 or 2
  const int lo = m < 8;
  const int r  = lo ? m : (m - 8);
  v2f A, A2;
  A[0]  = lo ? p1[r * 4 + kb]     : p2[r * 4 + kb];
  A[1]  = lo ? p1[r * 4 + kb + 1] : p2[r * 4 + kb + 1];
  A2[0] = lo ? p3[r * 4 + kb]     : 0.0f;   // S3 partials in rows 0..7
  A2[1] = lo ? p3[r * 4 + kb + 1] : 0.0f;
  v2f B;
  B[0] = 1.0f;
  B[1] = 1.0f;
  v8f C = {};
  v8f D1 = __builtin_amdgcn_wmma_f32_16x16x4_f32(
      false, A, false, B, (short)0, C, false, false);
  v8f D2 = __builtin_amdgcn_wmma_f32_16x16x4_f32(
      false, A2, false, B, (short)0, C, false, false);

  float r1 = ((D1[0] + D1[1]) + (D1[2] + D1[3])) +
             ((D1[4] + D1[5]) + (D1[6] + D1[7]));
  float r2 = ((D2[0] + D2[1]) + (D2[2] + D2[3])) +
             ((D2[4] + D2[5]) + (D2[6] + D2[7]));

  float S1 = __shfl(r1, 0, 32);
  float S2 = __shfl(r1, 16, 32);
  float S3 = __shfl(r2, 0, 32);

  if (tid == 0) {
    float invN = 1.0f / (float)N;
    float mu   = S1 * invN;
    out[0] = 2.0f * (S2 * invN - mu * mu) + S3 * invN;
  }
}

extern "C" void kernel_launch(void* const* d_in, const int* in_sizes, int n_in,
                              void* d_out, int out_size, void* d_ws, size_t ws_size,
                              hipStream_t stream) {
  const float* data   = (const float*)d_in[0];
  const float* target = (const float*)d_in[1];
  float* out = (float*)d_out;
  const int N = in_sizes[0];  // 16384
  matchloss_kernel<<<1, 1024, 0, stream>>>(data, target, out, N);
}